// ChamferDistance_8117488190299
// MI455X (gfx1250) — compile-verified
//
#include <hip/hip_runtime.h>

typedef __attribute__((ext_vector_type(2))) float v2f;
typedef __attribute__((ext_vector_type(8))) float v8f;

#define WAVES_PER_BLOCK 8

// One wave handles a 16-row tile of A-points; loops over all B-points in
// 16-column tiles using V_WMMA_F32_16X16X4_F32 to form
//   D[m][n] = qsq[n] - 2 * (p_m . q_n)
// The qsq[n] term rides in the K=3 padding slot (A[m][3]=1, B[3][n]=qsq[n]),
// so the accumulator input C is a loop-invariant zero vector (no per-tile
// broadcast movs). Running per-lane min is kept in the native 16x16 f32 D
// layout; ||p_m||^2 is added after the cross-lane min reduction. Partial sums
// are pre-scaled and atomically accumulated into accum[0].
__global__ void __launch_bounds__(256)
chamfer_dir_kernel(const float* __restrict__ Apts, const float* __restrict__ Bpts,
                   int M, int N, float scale, float* __restrict__ accum)
{
    const int b      = blockIdx.y;
    const int lane   = threadIdx.x & 31;
    const int wave   = threadIdx.x >> 5;
    const int lane16 = lane & 15;
    const bool hi    = lane >= 16;

    const float* Ab = Apts + (size_t)b * M * 3;
    const float* Bb = Bpts + (size_t)b * N * 3;

    // Load this wave's 16 A-points (each point duplicated in both halves).
    const int i0 = (blockIdx.x * WAVES_PER_BLOCK + wave) * 16;
    const int i  = i0 + lane16;
    const float px = Ab[i * 3 + 0];
    const float py = Ab[i * 3 + 1];
    const float pz = Ab[i * 3 + 2];
    const float psq = px * px + py * py + pz * pz;

    // A-matrix (16x4) VGPR layout: V0 = {K0 | K2}, V1 = {K1 | K3}
    // K0..K2 = -2*pred coords, K3 = 1.0 (multiplies qsq[n] carried in B).
    v2f a;
    a.x = hi ? (-2.0f * pz) : (-2.0f * px);
    a.y = hi ? 1.0f         : (-2.0f * py);

    // Loop-invariant zero accumulator input.
    v8f czero;
#pragma unroll
    for (int r = 0; r < 8; ++r) czero[r] = 0.0f;

    v8f runmin;
#pragma unroll
    for (int r = 0; r < 8; ++r) runmin[r] = 3.0e38f;

    for (int j0 = 0; j0 < N; j0 += 16) {
        const int j = j0 + lane16;
        const float tx = Bb[j * 3 + 0];
        const float ty = Bb[j * 3 + 1];
        const float tz = Bb[j * 3 + 2];
        const float qsq = tx * tx + ty * ty + tz * tz;

        // B-matrix (4x16) VGPR layout: V0 = {K0 | K2}, V1 = {K1 | K3}
        // K0..K2 = target coords, K3 = qsq[n].
        v2f bb;
        bb.x = hi ? tz  : tx;
        bb.y = hi ? qsq : ty;

        // D = A*B + 0 = qsq[n] - 2*cross[m][n]
        v8f d = __builtin_amdgcn_wmma_f32_16x16x4_f32(
            /*neg_a=*/false, a, /*neg_b=*/false, bb,
            /*c_mod=*/(short)0, czero, /*reuse_a=*/false, /*reuse_b=*/false);

#pragma unroll
        for (int r = 0; r < 8; ++r) runmin[r] = fminf(runmin[r], d[r]);
    }

    // Min across the 16 lanes of each half (rows r in lanes 0-15, r+8 in 16-31).
#pragma unroll
    for (int mask = 8; mask >= 1; mask >>= 1) {
#pragma unroll
        for (int r = 0; r < 8; ++r)
            runmin[r] = fminf(runmin[r], __shfl_xor(runmin[r], mask, 32));
    }

    // Add ||p_row||^2 and sum the 16 rows of this tile.
    float acc = 0.0f;
#pragma unroll
    for (int r = 0; r < 8; ++r) {
        const float psq_r = __shfl(psq, r + (hi ? 8 : 0), 32);  // row r / r+8
        acc += runmin[r] + psq_r;
    }
    acc += __shfl_xor(acc, 16, 32);   // combine rows 0-7 and 8-15

    if (lane == 0) atomicAdd(accum, acc * scale);
}

__global__ void chamfer_init_kernel(float* __restrict__ accum)
{
    if (threadIdx.x == 0 && blockIdx.x == 0) accum[0] = 0.0f;
}

__global__ void chamfer_finalize_kernel(const float* __restrict__ accum,
                                        float* __restrict__ out)
{
    if (threadIdx.x == 0 && blockIdx.x == 0) out[0] = accum[0];
}

extern "C" void kernel_launch(void* const* d_in, const int* in_sizes, int n_in,
                              void* d_out, int out_size, void* d_ws, size_t ws_size,
                              hipStream_t stream)
{
    const float* pred   = (const float*)d_in[0];
    const float* target = (const float*)d_in[1];
    float* out   = (float*)d_out;
    float* accum = (float*)d_ws;

    const int B = 4;
    const int M = in_sizes[0] / (B * 3);   // 8192
    const int N = in_sizes[1] / (B * 3);   // 8192

    chamfer_init_kernel<<<1, 32, 0, stream>>>(accum);

    dim3 block(32 * WAVES_PER_BLOCK);

    // Direction 1: for each pred point, min over targets. Mean weight 1/(B*M).
    dim3 grid1(M / (16 * WAVES_PER_BLOCK), B);
    chamfer_dir_kernel<<<grid1, block, 0, stream>>>(
        pred, target, M, N, 1.0f / (float)(B * M), accum);

    // Direction 2: for each target point, min over preds. Mean weight 1/(B*N).
    dim3 grid2(N / (16 * WAVES_PER_BLOCK), B);
    chamfer_dir_kernel<<<grid2, block, 0, stream>>>(
        target, pred, N, M, 1.0f / (float)(B * N), accum);

    chamfer_finalize_kernel<<<1, 32, 0, stream>>>(accum, out);
}